// MoEFeedForward_29583734735606
// MI455X (gfx1250) — compile-verified
//
#include <hip/hip_runtime.h>
#include <hip/hip_bf16.h>
#include <math.h>

#define Hdim 1024
#define Fdim 2048
#define Edim 8
#define Tdim 8192
#define CAP  2048
#define EPSV 1e-6f

typedef float v2f __attribute__((ext_vector_type(2)));
typedef float v8f __attribute__((ext_vector_type(8)));

// ---------------- kernel 1: RMSNorm + router gate (one wave per token) ----------------
__global__ __launch_bounds__(256) void k_rmsnorm_gate(
    const float* __restrict__ x, const float* __restrict__ rms_w,
    const float* __restrict__ gate_w, float* __restrict__ xn,
    int* __restrict__ topk_idx, float* __restrict__ topk_w)
{
  const int wave = threadIdx.x >> 5;
  const int lane = threadIdx.x & 31;
  const int t = blockIdx.x * 8 + wave;
  const float* xr = x + (size_t)t * Hdim;

  float xv[32];
  float ss = 0.f;
#pragma unroll
  for (int i = 0; i < 32; ++i) { float v = xr[i * 32 + lane]; xv[i] = v; ss += v * v; }
#pragma unroll
  for (int off = 16; off; off >>= 1) ss += __shfl_xor(ss, off, 32);
  const float r  = sqrtf(ss * (1.0f / Hdim) + EPSV);
  const float ir = 1.0f / r;

  float lg[Edim];
#pragma unroll
  for (int e = 0; e < Edim; ++e) lg[e] = 0.f;
#pragma unroll
  for (int i = 0; i < 32; ++i) {
    const int j = i * 32 + lane;
    const float v = xv[i] * ir * rms_w[j];
    xn[(size_t)t * Hdim + j] = v;
#pragma unroll
    for (int e = 0; e < Edim; ++e) lg[e] += v * gate_w[e * Hdim + j];
  }
#pragma unroll
  for (int e = 0; e < Edim; ++e) {
#pragma unroll
    for (int off = 16; off; off >>= 1) lg[e] += __shfl_xor(lg[e], off, 32);
  }

  if (lane == 0) {
    float mx = lg[0];
#pragma unroll
    for (int e = 1; e < Edim; ++e) mx = fmaxf(mx, lg[e]);
    float ex[Edim]; float sum = 0.f;
#pragma unroll
    for (int e = 0; e < Edim; ++e) { ex[e] = __expf(lg[e] - mx); sum += ex[e]; }
    int i0 = 0; float v0 = ex[0];
#pragma unroll
    for (int e = 1; e < Edim; ++e) if (ex[e] > v0) { v0 = ex[e]; i0 = e; }
    int i1 = (i0 == 0) ? 1 : 0; float v1 = ex[i1];
#pragma unroll
    for (int e = 0; e < Edim; ++e) if (e != i0 && ex[e] > v1) { v1 = ex[e]; i1 = e; }
    const float s0 = v0 / sum, s1 = v1 / sum;
    const float inv = 1.0f / (s0 + s1 + 1e-20f);
    topk_idx[2 * t] = i0;        topk_idx[2 * t + 1] = i1;
    topk_w[2 * t]   = s0 * inv;  topk_w[2 * t + 1]   = s1 * inv;
  }
}

// ---------------- kernel 2: deterministic capacity-limited dispatch ----------------
__global__ __launch_bounds__(256) void k_dispatch(
    const int* __restrict__ tki, const float* __restrict__ tkw,
    int* __restrict__ token_list, float* __restrict__ w_ec,
    int* __restrict__ slot_a)
{
  const int e = threadIdx.x >> 5;
  const int lane = threadIdx.x & 31;
  int cnt = 0;
  for (int base = 0; base < Tdim * 2; base += 32) {
    const int a  = base + lane;
    const int fe = tki[a];
    const unsigned m = (unsigned)__ballot(fe == e);
    const int prefix = __popc(m & ((1u << lane) - 1u));
    if (fe == e) {
      const int slot = cnt + prefix;
      if (slot < CAP) {
        token_list[e * CAP + slot] = a >> 1;
        w_ec[e * CAP + slot] = tkw[a];
        slot_a[a] = slot;
      } else {
        slot_a[a] = -1;
      }
    }
    cnt += __popc(m);
  }
  for (int s = cnt + lane; s < CAP; s += 32) {
    token_list[e * CAP + s] = 0;
    w_ec[e * CAP + s] = 0.f;
  }
}

// ---------------- async global -> LDS copy (CDNA5, tracked by ASYNCcnt) ----------------
__device__ __forceinline__ void async_copy_b128(unsigned ldsAddr, const float* base, unsigned voff) {
  asm volatile("global_load_async_to_lds_b128 %0, %1, %2"
               :: "v"(ldsAddr), "v"(voff), "s"(base)
               : "memory");
}

// ---------------- WMMA fp32 GEMM: out = X(gathered rows) * W^T ----------------
// Double-buffered LDS fed by async-to-LDS loads; V_WMMA_F32_16X16X4_F32 compute.
// MODE 0: plain store. MODE 1: dual-B, store silu(acc1)*acc2. MODE 2: store acc1*rowscale[row].
#define BM 128
#define BN 128
#define BK 32
#define LDSS 36                     // padded row stride (floats): 144B rows, conflict-free
#define TILEF (BM * LDSS)           // floats per LDS tile
#define TILEB (TILEF * 4)           // bytes per LDS tile

template <int MODE>
__global__ __launch_bounds__(256) void k_gemm_wmma(
    const float* __restrict__ X, long long x_es, int ldx,
    const int* __restrict__ rowmap, int rm_es,
    const float* __restrict__ W1, const float* __restrict__ W2, long long w_es,
    float* __restrict__ Out, long long o_es, int ldo,
    const float* __restrict__ rowscale, int rs_es,
    int K)
{
  // layout: As[0], As[1], Bs1[0], Bs1[1], (Bs2[0], Bs2[1])
  extern __shared__ float smem[];
  const unsigned smemBase = (unsigned)(uintptr_t)smem;   // LDS byte offset (flat low bits)

  const int e     = blockIdx.z;
  const int mBase = blockIdx.y * BM;
  const int nBase = blockIdx.x * BN;

  const float* Xe  = X  + (size_t)e * x_es;
  const float* W1e = W1 + (size_t)e * w_es;
  const float* W2e = (MODE == 1) ? (W2 + (size_t)e * w_es) : nullptr;
  float*       Oe  = Out + (size_t)e * o_es;
  const int*   rme = rowmap ? (rowmap + (size_t)e * rm_es) : nullptr;

  const int tid  = threadIdx.x;
  const int lane = tid & 31;
  const int wave = tid >> 5;
  const int mw   = wave >> 2;   // 0..1  (64-row slab)
  const int nw   = wave & 3;    // 0..3  (32-col slab)

  // per-thread load descriptors: 4 passes x float4
  unsigned aoff[4], boff[4], loff[4];
#pragma unroll
  for (int p = 0; p < 4; ++p) {
    const int linear = p * 256 + tid;
    const int row = linear >> 3;            // 8 float4 per 32-float row
    const int c4  = (linear & 7) * 4;
    const int grow = mBase + row;
    const int xr = rme ? rme[grow] : grow;
    aoff[p] = ((unsigned)xr * (unsigned)ldx + (unsigned)c4) * 4u;
    boff[p] = ((unsigned)(nBase + row) * (unsigned)K + (unsigned)c4) * 4u;
    loff[p] = (unsigned)(row * LDSS + c4) * 4u;
  }

  auto issue = [&](int buf, int k0) {
    const unsigned kb = (unsigned)k0 * 4u;
#pragma unroll
    for (int p = 0; p < 4; ++p) {
      async_copy_b128(smemBase + (unsigned)buf * TILEB + loff[p], Xe, aoff[p] + kb);
      async_copy_b128(smemBase + (unsigned)(2 + buf) * TILEB + loff[p], W1e, boff[p] + kb);
      if (MODE == 1)
        async_copy_b128(smemBase + (unsigned)(4 + buf) * TILEB + loff[p], W2e, boff[p] + kb);
    }
  };

  v8f acc1[4][2];
  v8f acc2[4][2];
#pragma unroll
  for (int i = 0; i < 4; ++i)
#pragma unroll
    for (int j = 0; j < 2; ++j) {
      acc1[i][j] = (v8f)0.0f;
      if (MODE == 1) acc2[i][j] = (v8f)0.0f;
    }

  const int hi = lane >> 4;      // 0/1 : selects K pair
  const int lr = lane & 15;      // row-within-tile

  issue(0, 0);
  int cur = 0;

  for (int k0 = 0; k0 < K; k0 += BK) {
    if (k0 + BK < K) {
      issue(1 - cur, k0 + BK);
      // async loads complete in order: <=NLOADS outstanding => current buffer resident
      asm volatile("s_wait_asynccnt %0" :: "n"(MODE == 1 ? 12 : 8) : "memory");
    } else {
      asm volatile("s_wait_asynccnt 0" ::: "memory");
    }
    __syncthreads();

    const float* Asb = smem + (size_t)cur * TILEF;
    const float* B1b = smem + (size_t)(2 + cur) * TILEF;
    const float* B2b = smem + (size_t)(4 + cur) * TILEF;

#pragma unroll
    for (int kk = 0; kk < BK; kk += 4) {
      v2f a[4], b1[2], b2[2];
#pragma unroll
      for (int i = 0; i < 4; ++i)
        a[i] = *(const v2f*)&Asb[(mw * 64 + i * 16 + lr) * LDSS + kk + 2 * hi];
#pragma unroll
      for (int j = 0; j < 2; ++j) {
        b1[j] = *(const v2f*)&B1b[(nw * 32 + j * 16 + lr) * LDSS + kk + 2 * hi];
        if (MODE == 1)
          b2[j] = *(const v2f*)&B2b[(nw * 32 + j * 16 + lr) * LDSS + kk + 2 * hi];
      }
#pragma unroll
      for (int i = 0; i < 4; ++i) {
#pragma unroll
        for (int j = 0; j < 2; ++j) {
          acc1[i][j] = __builtin_amdgcn_wmma_f32_16x16x4_f32(
              false, a[i], false, b1[j], (short)0, acc1[i][j], false, false);
          if (MODE == 1)
            acc2[i][j] = __builtin_amdgcn_wmma_f32_16x16x4_f32(
                false, a[i], false, b2[j], (short)0, acc2[i][j], false, false);
        }
      }
    }
    __syncthreads();
    cur ^= 1;
  }

  // ---- epilogue: 16x16 f32 C/D layout: n = lane&15, m = v + 8*(lane>>4) ----
#pragma unroll
  for (int i = 0; i < 4; ++i) {
#pragma unroll
    for (int j = 0; j < 2; ++j) {
#pragma unroll
      for (int v = 0; v < 8; ++v) {
        const int m = mw * 64 + i * 16 + hi * 8 + v;     // local row 0..127
        const int n = nBase + nw * 32 + j * 16 + lr;     // global col
        float val;
        if (MODE == 1) {
          const float g = acc1[i][j][v];
          const float u = acc2[i][j][v];
          val = (g / (1.0f + __expf(-g))) * u;           // silu(g)*u
        } else if (MODE == 2) {
          val = acc1[i][j][v] * rowscale[(size_t)e * rs_es + mBase + m];
        } else {
          val = acc1[i][j][v];
        }
        Oe[(size_t)(mBase + m) * ldo + n] = val;
      }
    }
  }
}

// ---------------- kernel 7: combine shared + routed contributions ----------------
__global__ __launch_bounds__(256) void k_combine(
    const float* __restrict__ ysh, const float* __restrict__ ye,
    const int* __restrict__ tki, const int* __restrict__ slot_a,
    float* __restrict__ out)
{
  const int t  = blockIdx.x;
  const int c  = threadIdx.x;                  // 256 float4 per row (H=1024)
  const int e0 = tki[2 * t],     e1 = tki[2 * t + 1];
  const int s0 = slot_a[2 * t],  s1 = slot_a[2 * t + 1];

  float4 a = ((const float4*)ysh)[(size_t)t * 256 + c];
  if (s0 >= 0) {
    const float4 b = ((const float4*)ye)[((size_t)e0 * CAP + s0) * 256 + c];
    a.x += b.x; a.y += b.y; a.z += b.z; a.w += b.w;
  }
  if (s1 >= 0) {
    const float4 b = ((const float4*)ye)[((size_t)e1 * CAP + s1) * 256 + c];
    a.x += b.x; a.y += b.y; a.z += b.z; a.w += b.w;
  }
  ((float4*)out)[(size_t)t * 256 + c] = a;
  if (t == 0 && c == 0) out[(size_t)Tdim * Hdim] = 0.0f;   // aux_loss (eval mode)
}

// ---------------- host launcher ----------------
extern "C" void kernel_launch(void* const* d_in, const int* in_sizes, int n_in,
                              void* d_out, int out_size, void* d_ws, size_t ws_size,
                              hipStream_t stream) {
  const float* x      = (const float*)d_in[0];
  const float* rms_w  = (const float*)d_in[1];
  const float* gate_w = (const float*)d_in[2];
  const float* wg     = (const float*)d_in[3];
  const float* wu     = (const float*)d_in[4];
  const float* wd     = (const float*)d_in[5];
  const float* swg    = (const float*)d_in[6];
  const float* swu    = (const float*)d_in[7];
  const float* swd    = (const float*)d_in[8];
  float* out = (float*)d_out;

  char* w = (char*)d_ws;
  float* xn         = (float*)w; w += (size_t)Tdim * Hdim * 4;          // 33.5 MB
  float* ysh        = (float*)w; w += (size_t)Tdim * Hdim * 4;          // 33.5 MB
  float* act        = (float*)w; w += (size_t)Edim * CAP * Fdim * 4;    // 134 MB (reused by shared act)
  float* ye         = (float*)w; w += (size_t)Edim * CAP * Hdim * 4;    // 67 MB
  int*   tki        = (int*)w;   w += (size_t)Tdim * 2 * 4;
  float* tkw        = (float*)w; w += (size_t)Tdim * 2 * 4;
  int*   token_list = (int*)w;   w += (size_t)Edim * CAP * 4;
  float* w_ec       = (float*)w; w += (size_t)Edim * CAP * 4;
  int*   slot_a     = (int*)w;   w += (size_t)Tdim * 2 * 4;

  const dim3 blk(256);
  const size_t smem_dual   = (size_t)6 * TILEB;   // A x2, B1 x2, B2 x2
  const size_t smem_single = (size_t)4 * TILEB;   // A x2, B1 x2

  // 1) RMSNorm + gate
  k_rmsnorm_gate<<<Tdim / 8, blk, 0, stream>>>(x, rms_w, gate_w, xn, tki, tkw);

  // 2) Deterministic dispatch
  k_dispatch<<<1, blk, 0, stream>>>(tki, tkw, token_list, w_ec, slot_a);

  // 3) Shared expert GEMM1: act = silu(xn@swg^T) * (xn@swu^T)   [T x F]
  k_gemm_wmma<1><<<dim3(Fdim / BN, Tdim / BM, 1), blk, smem_dual, stream>>>(
      xn, 0, Hdim, nullptr, 0, swg, swu, 0, act, 0, Fdim, nullptr, 0, Hdim);

  // 4) Shared expert GEMM2: ysh = act @ swd^T                   [T x H]
  k_gemm_wmma<0><<<dim3(Hdim / BN, Tdim / BM, 1), blk, smem_single, stream>>>(
      act, 0, Fdim, nullptr, 0, swd, nullptr, 0, ysh, 0, Hdim, nullptr, 0, Fdim);

  // 5) Routed GEMM1 (gathered rows): act[e] = silu(Xg@wg[e]^T)*(Xg@wu[e]^T)  [E x C x F]
  k_gemm_wmma<1><<<dim3(Fdim / BN, CAP / BM, Edim), blk, smem_dual, stream>>>(
      xn, 0, Hdim, token_list, CAP, wg, wu, (long long)Fdim * Hdim,
      act, (long long)CAP * Fdim, Fdim, nullptr, 0, Hdim);

  // 6) Routed GEMM2 + per-row routing weight: ye[e] = (act[e] @ wd[e]^T) * w_ec  [E x C x H]
  k_gemm_wmma<2><<<dim3(Hdim / BN, CAP / BM, Edim), blk, smem_single, stream>>>(
      act, (long long)CAP * Fdim, Fdim, nullptr, 0, wd, nullptr, (long long)Hdim * Fdim,
      ye, (long long)CAP * Hdim, Hdim, w_ec, CAP, Fdim);

  // 7) Combine (deterministic gather, no atomics) + aux loss
  k_combine<<<Tdim, blk, 0, stream>>>(ysh, ye, tki, slot_a, out);
}